// LGNLayer_10127532884487
// MI455X (gfx1250) — compile-verified
//
#include <hip/hip_runtime.h>
#include <cstdint>

// ---------------------------------------------------------------------------
// LGN layer forward for MI455X (gfx1250).
//  N = 8192. Inputs (setup_inputs order):
//   0: x[N] (unused)  1: is_firing[N]  2: node_weights[N*N]
//   3: node_thresholds[N]  4: lgn_weights[N*N]  5: lgn_threshold[N]
//  Output: firing[N] ++ lw[N*N] ++ th[N]  (fp32)
//
// Bandwidth roofline: ~1 GB HBM traffic (2 matvec reads + lw copy) -> ~44 us
// at 23.3 TB/s; FLOPs are negligible, so fp32 WMMA 16x16x4 is used for the
// matvec accumulation (precision preserved, matrix pipe engaged, zero cost).
// Pipeline: stage s+1 tile loads issue before stage s's WMMA compute; the
// tile ping-pongs between two LDS buffers (register pressure stays ~80 VGPRs,
// no scratch spills).
// ---------------------------------------------------------------------------

#define NN      8192
#define KPART   2048          // K range per wave (4 parts per 16-row tile)
#define KCHUNK  64            // K columns staged per LDS tile
#define NSTAGE  (KPART / KCHUNK)          // 32
#define TSTRIDE 68            // LDS row stride (floats); 68 % 64 == 4 -> A reads bank-conflict-free
#define NLD     (16 * (KCHUNK / 4) / 32)  // float4 loads per lane per stage = 8

typedef float v2f  __attribute__((ext_vector_type(2)));
typedef float v8f  __attribute__((ext_vector_type(8)));
typedef unsigned int u32x4 __attribute__((ext_vector_type(4)));
typedef int i32x8 __attribute__((ext_vector_type(8)));
typedef int i32x4 __attribute__((ext_vector_type(4)));

// ---------------------------------------------------------------------------
// Kernel 1/3: partial mat-vec, WMMA f32 16x16x4.
//  grid = (NN/16 tiles) * 2 blocks, 64 threads (2 waves) per block.
//  wave handles rows [tile*16, tile*16+16), K range [part*KPART, ...).
//  Writes partials[part*NN + row] (to be summed over 4 parts).
// ---------------------------------------------------------------------------
__global__ __launch_bounds__(64) void matvec_wmma(const float* __restrict__ W,
                                                  const float* __restrict__ vec,
                                                  float* __restrict__ partials) {
  __shared__ float vecsh[2][KPART];              // 16 KB: per-wave vector slice
  __shared__ float tilesh[2][2][16 * TSTRIDE];   // 17.4 KB: per-wave double-buffered W tile

  const int wave = threadIdx.x >> 5;
  const int lane = threadIdx.x & 31;
  const int tile = blockIdx.x >> 1;
  const int part = ((blockIdx.x & 1) << 1) | wave;
  const int row0 = tile * 16;
  const int kbase = part * KPART;

  // ---- TDM async load of this wave's 8 KB vector slice into LDS ----------
  {
    unsigned lds_off = (unsigned)(uintptr_t)(&vecsh[wave][0]);
    unsigned long long ga = (unsigned long long)(uintptr_t)(vec + kbase);
    // D# group0: count=1 | lds_addr | global_addr[56:0] | type=2
    u32x4 g0 = { 1u,
                 lds_off,
                 (unsigned)(ga & 0xFFFFFFFFull),
                 (unsigned)((ga >> 32) & 0x01FFFFFFull) | 0x80000000u };
    // D# group1: data_size=4B; tensor_dim0=KPART, tensor_dim1=1;
    //            tile_dim0=KPART, tile_dim1=1; dim0_stride=KPART.
    i32x8 g1 = { 0x00020000,                       // data_size=2 (4 bytes)
                 (int)((KPART & 0xFFFF) << 16),    // tensor_dim0 lo16
                 (int)(1 << 16),                   // tensor_dim0 hi16=0 | tensor_dim1 lo16=1
                 (int)(KPART << 16),               // tensor_dim1 hi16=0 | tile_dim0
                 1,                                // tile_dim1=1, tile_dim2=0
                 KPART,                            // tensor_dim0_stride lo32
                 0, 0 };
    i32x4 z4 = { 0, 0, 0, 0 };
    i32x8 z8 = { 0, 0, 0, 0, 0, 0, 0, 0 };
    __builtin_amdgcn_tensor_load_to_lds(g0, g1, z4, z4, z8, 0);
  }

  // per-lane stage geometry: q = i*32 + lane ; r = q/16 ; c4 = q%16
  // -> lanes sweep 16 consecutive float4 (256 B) of a row: fully coalesced.
  const float* Wbase = W + (size_t)(row0 + (lane >> 4)) * NN + kbase + (lane & 15) * 4;
  float* ldsbase0 = &tilesh[wave][0][(lane >> 4) * TSTRIDE + (lane & 15) * 4];
  float* ldsbase1 = &tilesh[wave][1][(lane >> 4) * TSTRIDE + (lane & 15) * 4];

  // ---- prologue: stage-0 tile -> regs -> LDS buf0 (overlaps TDM copy) ----
  float4 buf[NLD];
#pragma unroll
  for (int i = 0; i < NLD; ++i)
    buf[i] = *(const float4*)(Wbase + (size_t)(2 * i) * NN);
#pragma unroll
  for (int i = 0; i < NLD; ++i)
    *(float4*)(ldsbase0 + 2 * i * TSTRIDE) = buf[i];

  __builtin_amdgcn_s_wait_tensorcnt(0);

  v8f acc = {};
  const int h = lane >> 4;          // lane half selects K sub-pair per WMMA layout
  const int m = lane & 15;          // A-matrix row within tile

  for (int s = 0; s < NSTAGE; ++s) {
    // issue next stage's global loads first; they ride out the compute below
    const int sn = (s + 1 < NSTAGE) ? s + 1 : s;
#pragma unroll
    for (int i = 0; i < NLD; ++i)
      buf[i] = *(const float4*)(Wbase + (size_t)(2 * i) * NN + sn * KCHUNK);

    // compute: A from LDS tile (buffer s&1), B = broadcast vector slice
    const float* arow = &tilesh[wave][s & 1][m * TSTRIDE];
    const float* vrow = &vecsh[wave][s * KCHUNK];
#pragma unroll
    for (int kk = 0; kk < KCHUNK; kk += 4) {
      // A (16x4): lane(m,h), vgpr j -> A[m][2h+j] ; B (4x16) broadcast of vec:
      // B[k][n] = v[k] for all n, same (h,j)->k striping.
      v2f a, b;
      a.x = arow[kk + 2 * h];  a.y = arow[kk + 2 * h + 1];
      b.x = vrow[kk + 2 * h];  b.y = vrow[kk + 2 * h + 1];
      acc = __builtin_amdgcn_wmma_f32_16x16x4_f32(false, a, false, b,
                                                  (short)0, acc, false, false);
    }

    // commit prefetched tile into the other LDS buffer (loads have landed;
    // same-wave DS ordering keeps the ping-pong race-free without barriers)
    float* ldst = ((s + 1) & 1) ? ldsbase1 : ldsbase0;
#pragma unroll
    for (int i = 0; i < NLD; ++i)
      *(float4*)(ldst + 2 * i * TSTRIDE) = buf[i];
  }

  // D column n=0 lives in lanes 0 (rows 0..7 in acc[0..7]) and 16 (rows 8..15)
  if (m == 0) {
    const int rbase = row0 + h * 8;
#pragma unroll
    for (int r = 0; r < 8; ++r) partials[part * NN + rbase + r] = acc[r];
  }
}

// ---------------------------------------------------------------------------
// Kernel 2: sum 4 partials, threshold -> firing (written straight to d_out)
// ---------------------------------------------------------------------------
__global__ __launch_bounds__(256) void fire_kernel(const float* __restrict__ partials,
                                                   const float* __restrict__ thr,
                                                   float* __restrict__ firing_out) {
  const int i = blockIdx.x * 256 + threadIdx.x;
  const float s = partials[i] + partials[NN + i] + partials[2 * NN + i] + partials[3 * NN + i];
  firing_out[i] = (s > thr[i]) ? 1.0f : 0.0f;
}

// ---------------------------------------------------------------------------
// Kernel 4: single block. yAct + argmax, then precompute the updated column,
// scaled row, thresholds-out, and scalar (mi, ma, fire).
// ---------------------------------------------------------------------------
__global__ __launch_bounds__(1024) void finalize_k(const float* __restrict__ partials,
                                                   const float* __restrict__ lgn_thr,
                                                   const float* __restrict__ lgnW,
                                                   const float* __restrict__ firing,
                                                   float* __restrict__ thr_out,
                                                   float* __restrict__ col_new,
                                                   float* __restrict__ row_scaled,
                                                   float* __restrict__ scal) {
  __shared__ float sv[1024];
  __shared__ int   si[1024];
  __shared__ float s_ma; __shared__ int s_mi; __shared__ int s_fire;

  float best = -3.402823e38f; int bidx = NN;
  for (int i = threadIdx.x; i < NN; i += 1024) {
    float y1 = partials[i] + partials[NN + i] + partials[2 * NN + i] + partials[3 * NN + i];
    y1 = (y1 < 0.0f) ? 0.9f : y1;
    const float ya = fmaxf(y1 - lgn_thr[i], 0.0f);
    if (ya > best) { best = ya; bidx = i; }   // ascending i -> first max kept
  }
  sv[threadIdx.x] = best; si[threadIdx.x] = bidx;
  __syncthreads();
  for (int off = 512; off > 0; off >>= 1) {
    if (threadIdx.x < off) {
      const float ov = sv[threadIdx.x + off]; const int oi = si[threadIdx.x + off];
      if (ov > sv[threadIdx.x] || (ov == sv[threadIdx.x] && oi < si[threadIdx.x])) {
        sv[threadIdx.x] = ov; si[threadIdx.x] = oi;
      }
    }
    __syncthreads();
  }
  if (threadIdx.x == 0) {
    s_ma = sv[0]; s_mi = si[0]; s_fire = (sv[0] > 0.0f) ? 1 : 0;
    scal[0] = (float)si[0]; scal[1] = sv[0]; scal[2] = (sv[0] > 0.0f) ? 1.0f : 0.0f;
  }
  __syncthreads();
  const int mi = s_mi; const float ma = s_ma; const int fire = s_fire;

  for (int j = threadIdx.x; j < NN; j += 1024) {
    col_new[j] = fire ? (lgnW[(size_t)mi * NN + j] + 0.1f * ma * firing[j])
                      : lgnW[(size_t)j * NN + mi];
    thr_out[j] = lgn_thr[j] + ((j == mi && fire) ? 0.005f * ma : 0.0f);
  }

  // mean of updated row mi (diagonal element got the column update first)
  float psum = 0.0f;
  for (int c = threadIdx.x; c < NN; c += 1024) {
    const float rv = (c == mi)
        ? (lgnW[(size_t)mi * NN + mi] + (fire ? 0.1f * ma * firing[mi] : 0.0f))
        : lgnW[(size_t)mi * NN + c];
    psum += rv;
  }
  sv[threadIdx.x] = psum;
  __syncthreads();
  for (int off = 512; off > 0; off >>= 1) {
    if (threadIdx.x < off) sv[threadIdx.x] += sv[threadIdx.x + off];
    __syncthreads();
  }
  const float scale = 2.5f * (float)NN / sv[0];

  for (int c = threadIdx.x; c < NN; c += 1024) {
    const float rv = (c == mi)
        ? (lgnW[(size_t)mi * NN + mi] + (fire ? 0.1f * ma * firing[mi] : 0.0f))
        : lgnW[(size_t)mi * NN + c];
    row_scaled[c] = fire ? rv * scale : lgnW[(size_t)mi * NN + c];
  }
}

// ---------------------------------------------------------------------------
// Kernel 5: stream lgn_weights -> lw output (float4), substituting the
// updated column mi and row mi. 512 MB of traffic -> dominant cost.
// ---------------------------------------------------------------------------
__global__ __launch_bounds__(256) void copy_update(const float* __restrict__ lgnW,
                                                   const float* __restrict__ col_new,
                                                   const float* __restrict__ row_scaled,
                                                   const float* __restrict__ scal,
                                                   float* __restrict__ lw_out) {
  const int t  = blockIdx.x * 256 + threadIdx.x;  // float4 index
  const int r  = t >> 11;                          // / (NN/4)
  const int c4 = t & 2047;
  float4 v = ((const float4*)lgnW)[t];
  const int mi = (int)scal[0];
  const int fire = (scal[2] != 0.0f);
  if (fire) {
    if (r == mi) {
      v = ((const float4*)row_scaled)[c4];
    } else if (c4 == (mi >> 2)) {
      ((float*)&v)[mi & 3] = col_new[r];
    }
  }
  ((float4*)lw_out)[t] = v;
}

// ---------------------------------------------------------------------------
extern "C" void kernel_launch(void* const* d_in, const int* in_sizes, int n_in,
                              void* d_out, int out_size, void* d_ws, size_t ws_size,
                              hipStream_t stream) {
  const float* is_firing = (const float*)d_in[1];
  const float* node_w    = (const float*)d_in[2];
  const float* node_thr  = (const float*)d_in[3];
  const float* lgn_w     = (const float*)d_in[4];
  const float* lgn_thr   = (const float*)d_in[5];

  float* out        = (float*)d_out;
  float* out_firing = out;
  float* out_lw     = out + NN;
  float* out_th     = out + NN + (size_t)NN * NN;

  float* ws         = (float*)d_ws;
  float* partials   = ws;            // 4*NN floats (reused by both matvecs)
  float* col_new    = ws + 4 * NN;   // NN
  float* row_scaled = ws + 5 * NN;   // NN
  float* scal       = ws + 6 * NN;   // mi, ma, fire

  const int mv_blocks = (NN / 16) * 2;  // 1024 blocks x 64 threads

  // retina matvec -> firing
  matvec_wmma<<<mv_blocks, 64, 0, stream>>>(node_w, is_firing, partials);
  fire_kernel<<<NN / 256, 256, 0, stream>>>(partials, node_thr, out_firing);
  // LGN matvec -> yAct / argmax / update precompute
  matvec_wmma<<<mv_blocks, 64, 0, stream>>>(lgn_w, out_firing, partials);
  finalize_k<<<1, 1024, 0, stream>>>(partials, lgn_thr, lgn_w, out_firing,
                                     out_th, col_new, row_scaled, scal);
  // bulk copy with row/column substitution
  copy_update<<<((size_t)NN * NN / 4) / 256, 256, 0, stream>>>(lgn_w, col_new,
                                                               row_scaled, scal, out_lw);
}